// PraxisAttention_15307263443043
// MI455X (gfx1250) — compile-verified
//
#include <hip/hip_runtime.h>
#include <hip/hip_bf16.h>

#define BATCH 2
#define SLEN  2048
#define DMODEL 1024
#define NHEAD 16
#define HDIM  64

typedef __attribute__((ext_vector_type(2))) float v2f;
typedef __attribute__((ext_vector_type(8))) float v8f;

__device__ __forceinline__ v8f wmma4(v2f a, v2f b, v8f c) {
  // D = A(16x4 f32) * B(4x16 f32) + C(16x16 f32)
  return __builtin_amdgcn_wmma_f32_16x16x4_f32(false, a, false, b, (short)0, c, false, false);
}

// Low 32 bits of a flat shared-memory address are the LDS byte offset
// (LDS aperture lives in addr[63:32] per the gfx1250 aperture check).
__device__ __forceinline__ uint32_t lds_off(const void* p) {
  return (uint32_t)(uintptr_t)p;
}

// ---------------------------------------------------------------------------
// lambda = exp(lq1.lk1) - exp(lq2.lk2) + 0.8   (one wave)
// ---------------------------------------------------------------------------
__global__ void lambda_kernel(const float* __restrict__ lq1, const float* __restrict__ lq2,
                              const float* __restrict__ lk1, const float* __restrict__ lk2,
                              float* __restrict__ lam) {
  int l = threadIdx.x; // 0..31
  float d1 = lq1[l] * lk1[l] + lq1[l + 32] * lk1[l + 32];
  float d2 = lq2[l] * lk2[l] + lq2[l + 32] * lk2[l + 32];
  #pragma unroll
  for (int off = 16; off > 0; off >>= 1) {
    d1 += __shfl_xor(d1, off, 32);
    d2 += __shfl_xor(d2, off, 32);
  }
  if (l == 0) lam[0] = __expf(d1) - __expf(d2) + 0.8f;
}

// ---------------------------------------------------------------------------
// fp32 WMMA GEMM with async global->LDS double-buffered staging.
// Block tile: 16 rows x 512 cols (8 waves x 64 cols). K staged in blocks of 16.
// Per K-block each wave issues exactly 9 async-copy instructions
// (8x b128 for B, 1x b32 for A) -> s_wait_asynccnt 9 gives double buffering.
// mode 0: plain row-major store
// mode 1: scatter to (B,H,S,128)  (q/k projections)
// mode 2: scatter to (B,H,S,64)   (v projection)
// ---------------------------------------------------------------------------
__global__ void gemm_wmma(const float* __restrict__ A, const float* __restrict__ B,
                          float* __restrict__ C, int M, int N, int K, int mode) {
  __shared__ float lA[2][16 * 18];    // padded stride 18
  __shared__ float lB[2][16 * 512];

  const int tid  = threadIdx.x;
  const int wave = tid >> 5;
  const int lane = tid & 31;
  const int half = lane >> 4;
  const int r    = lane & 15;

  const int blocks_n = N >> 9;              // 512-col block tiles per row
  const int tm  = blockIdx.x / blocks_n;
  const int nb  = blockIdx.x - tm * blocks_n;
  const int row0 = tm << 4;
  const int NC0  = nb << 9;
  const int wcol = wave << 6;

  auto prefetch = [&](int buf, int kb) {
    const int k0 = kb << 4;
    // A panel: 16x16, one b32 per thread (coalesced 64B per row)
    {
      const int arow = tid >> 4, ak = tid & 15;
      const float* g = A + (size_t)(row0 + arow) * K + k0 + ak;
      uint32_t l = lds_off(&lA[buf][arow * 18 + ak]);
      asm volatile("global_load_async_to_lds_b32 %0, %1, off" :: "v"(l), "v"(g) : "memory");
    }
    // B panel: 16x512, eight b128 per thread (fully coalesced)
    #pragma unroll
    for (int i = 0; i < 8; ++i) {
      const int c = tid + (i << 8);          // chunk 0..2047 (16B each)
      const int brow = c >> 7, bc4 = (c & 127) << 2;
      const float* g = B + (size_t)(k0 + brow) * N + NC0 + bc4;
      uint32_t l = lds_off(&lB[buf][brow * 512 + bc4]);
      asm volatile("global_load_async_to_lds_b128 %0, %1, off" :: "v"(l), "v"(g) : "memory");
    }
  };

  v8f acc[4] = {};
  const int KB = K >> 4;

  prefetch(0, 0);
  for (int kb = 0; kb < KB; ++kb) {
    const int buf = kb & 1;
    if (kb + 1 < KB) {
      prefetch(buf ^ 1, kb + 1);
      // current buffer's 9 ops are older than the 9 just issued -> done when cnt<=9
      asm volatile("s_wait_asynccnt 0x9" ::: "memory");
    } else {
      asm volatile("s_wait_asynccnt 0x0" ::: "memory");
    }
    __syncthreads();

    const float* bA = &lA[buf][r * 18 + 2 * half];
    const float* bB = &lB[buf][(2 * half) * 512 + wcol + r];
    #pragma unroll
    for (int ki = 0; ki < 4; ++ki) {
      v2f a = *(const v2f*)(bA + 4 * ki);
      const float* bp = bB + (size_t)ki * 4 * 512;
      #pragma unroll
      for (int nt = 0; nt < 4; ++nt) {
        v2f b;
        b[0] = bp[nt * 16];
        b[1] = bp[512 + nt * 16];
        acc[nt] = wmma4(a, b, acc[nt]);
      }
    }
    __syncthreads();   // all waves done reading `buf` before it is refilled
  }

  #pragma unroll
  for (int nt = 0; nt < 4; ++nt) {
    const int n = NC0 + wcol + nt * 16 + r;
    #pragma unroll
    for (int p = 0; p < 8; ++p) {
      const int m = row0 + p + 8 * half;
      const float v = acc[nt][p];
      if (mode == 0) {
        C[(size_t)m * N + n] = v;
      } else if (mode == 1) {
        const int b = m >> 11, s = m & (SLEN - 1);
        const int h = n >> 7, c = n & 127;
        C[(((size_t)(b * NHEAD + h) * SLEN + s) << 7) + c] = v;
      } else {
        const int b = m >> 11, s = m & (SLEN - 1);
        const int h = n >> 6, c = n & 63;
        C[(((size_t)(b * NHEAD + h) * SLEN + s) << 6) + c] = v;
      }
    }
  }
}

// ---------------------------------------------------------------------------
// Differential flash attention.
// Q,K: (B,H,S,128)  [cols 0..63 = branch1, 64..127 = branch2]
// V:   (B,H,S,64)   AO: (B,H,S,64)
// One wave owns one 16-row query tile; dual online softmax; causal early-exit.
// ---------------------------------------------------------------------------
__global__ void attn_kernel(const float* __restrict__ Qb, const float* __restrict__ Kb,
                            const float* __restrict__ Vb, const float* __restrict__ maskp,
                            const float* __restrict__ lamp, float* __restrict__ AO) {
  __shared__ float ldsP1[8][16 * 18];
  __shared__ float ldsP2[8][16 * 18];

  const int wave = threadIdx.x >> 5;
  const int lane = threadIdx.x & 31;
  const int half = lane >> 4;
  const int r    = lane & 15;

  const int bh   = blockIdx.x >> 4;   // 0..31
  const int qblk = blockIdx.x & 15;   // 0..15
  const int q0   = (qblk * 8 + wave) * 16;
  const int b    = bh >> 4;
  const int h    = bh & 15;

  const float slope = exp2f(-0.5f * (float)(h + 1));
  const float lam   = lamp[0];
  const float scale = 0.125f;  // 1/sqrt(64)

  v2f q1f[16], q2f[16];
  {
    const float* qrow = Qb + (((size_t)bh * SLEN + q0 + r) << 7) + 2 * half;
    #pragma unroll
    for (int j = 0; j < 16; ++j) {
      q1f[j] = *(const v2f*)(qrow + 4 * j);
      q2f[j] = *(const v2f*)(qrow + 64 + 4 * j);
    }
  }

  v8f o1[4] = {}, o2[4] = {};
  float m1[8], l1[8], m2[8], l2[8];
  #pragma unroll
  for (int p = 0; p < 8; ++p) { m1[p] = -1e30f; l1[p] = 0.f; m2[p] = -1e30f; l2[p] = 0.f; }

  const float* kbase = Kb + (((size_t)bh * SLEN) << 7);
  const float* vbase = Vb + (((size_t)bh * SLEN) << 6);
  const float* mrow  = maskp + (size_t)b * SLEN;

  const int nkt = (q0 >> 4) + 1;  // causal: key tiles 0 .. q0/16
  for (int kt = 0; kt < nkt; ++kt) {
    const int kv0 = kt << 4;

    v8f s1 = {}, s2 = {};
    {
      const float* krow = kbase + (((size_t)(kv0 + r)) << 7) + 2 * half;
      #pragma unroll
      for (int j = 0; j < 16; ++j) {
        v2f b1 = *(const v2f*)(krow + 4 * j);
        s1 = wmma4(q1f[j], b1, s1);
        v2f b2 = *(const v2f*)(krow + 64 + 4 * j);
        s2 = wmma4(q2f[j], b2, s2);
      }
    }

    const float kpos = (float)(kv0 + r);
    const float mb   = (1.0f - mrow[kv0 + r]) * -1e9f;
    #pragma unroll
    for (int p = 0; p < 8; ++p) {
      const float qpos = (float)(q0 + p + 8 * half);
      const float bias = -slope * (qpos - kpos) + ((kpos > qpos) ? -1e9f : 0.f) + mb;
      s1[p] = s1[p] * scale + bias;
      s2[p] = s2[p] * scale + bias;
    }

    float t1[8], t2[8];
    #pragma unroll
    for (int p = 0; p < 8; ++p) { t1[p] = s1[p]; t2[p] = s2[p]; }
    #pragma unroll
    for (int off = 1; off < 16; off <<= 1) {
      #pragma unroll
      for (int p = 0; p < 8; ++p) {
        t1[p] = fmaxf(t1[p], __shfl_xor(t1[p], off, 16));
        t2[p] = fmaxf(t2[p], __shfl_xor(t2[p], off, 16));
      }
    }

    float a1[8], a2[8], p1v[8], p2v[8];
    #pragma unroll
    for (int p = 0; p < 8; ++p) {
      const float mn1 = fmaxf(m1[p], t1[p]);
      a1[p]  = __expf(m1[p] - mn1);
      p1v[p] = __expf(s1[p] - mn1);
      m1[p]  = mn1;
      const float mn2 = fmaxf(m2[p], t2[p]);
      a2[p]  = __expf(m2[p] - mn2);
      p2v[p] = __expf(s2[p] - mn2);
      m2[p]  = mn2;
    }
    float r1[8], r2[8];
    #pragma unroll
    for (int p = 0; p < 8; ++p) { r1[p] = p1v[p]; r2[p] = p2v[p]; }
    #pragma unroll
    for (int off = 1; off < 16; off <<= 1) {
      #pragma unroll
      for (int p = 0; p < 8; ++p) {
        r1[p] += __shfl_xor(r1[p], off, 16);
        r2[p] += __shfl_xor(r2[p], off, 16);
      }
    }
    #pragma unroll
    for (int p = 0; p < 8; ++p) {
      l1[p] = l1[p] * a1[p] + r1[p];
      l2[p] = l2[p] * a2[p] + r2[p];
    }
    #pragma unroll
    for (int nt = 0; nt < 4; ++nt)
      #pragma unroll
      for (int p = 0; p < 8; ++p) {
        o1[nt][p] *= a1[p];
        o2[nt][p] *= a2[p];
      }

    // transpose P (C-layout -> A-layout) via wave-local LDS
    #pragma unroll
    for (int p = 0; p < 8; ++p) {
      ldsP1[wave][(p + 8 * half) * 18 + r] = p1v[p];
      ldsP2[wave][(p + 8 * half) * 18 + r] = p2v[p];
    }
    asm volatile("" ::: "memory");  // same-wave DS ops are in-order; block IR reordering

    #pragma unroll
    for (int j = 0; j < 4; ++j) {
      v2f ap1 = *(const v2f*)(&ldsP1[wave][r * 18 + 4 * j + 2 * half]);
      v2f ap2 = *(const v2f*)(&ldsP2[wave][r * 18 + 4 * j + 2 * half]);
      const float* vrow = vbase + (((size_t)(kv0 + 4 * j + 2 * half)) << 6) + r;
      #pragma unroll
      for (int nt = 0; nt < 4; ++nt) {
        v2f bv;
        bv[0] = vrow[nt * 16];
        bv[1] = vrow[64 + nt * 16];
        o1[nt] = wmma4(ap1, bv, o1[nt]);
        o2[nt] = wmma4(ap2, bv, o2[nt]);
      }
    }
    asm volatile("" ::: "memory");
  }

  float* aorow = AO + (((size_t)bh * SLEN + q0) << 6);
  #pragma unroll
  for (int nt = 0; nt < 4; ++nt) {
    #pragma unroll
    for (int p = 0; p < 8; ++p) {
      const float val = o1[nt][p] / l1[p] - lam * (o2[nt][p] / l2[p]);
      aorow[(size_t)(p + 8 * half) * 64 + nt * 16 + r] = val;
    }
  }
}

// ---------------------------------------------------------------------------
// Per-(b,h) mean / rstd over S*D elements of AO.
// ---------------------------------------------------------------------------
__global__ void stats_kernel(const float* __restrict__ AO, float* __restrict__ stats) {
  const int bh = blockIdx.x;
  const float* p = AO + ((size_t)bh * SLEN << 6);
  float s = 0.f, ss = 0.f;
  for (int i = threadIdx.x; i < SLEN * HDIM; i += blockDim.x) {
    const float v = p[i];
    s += v; ss += v * v;
  }
  #pragma unroll
  for (int off = 16; off > 0; off >>= 1) {
    s  += __shfl_xor(s, off, 32);
    ss += __shfl_xor(ss, off, 32);
  }
  __shared__ float sh[16];
  const int wave = threadIdx.x >> 5, lane = threadIdx.x & 31;
  if (lane == 0) { sh[wave] = s; sh[8 + wave] = ss; }
  __syncthreads();
  if (threadIdx.x == 0) {
    float S1 = 0.f, S2 = 0.f;
    #pragma unroll
    for (int w = 0; w < 8; ++w) { S1 += sh[w]; S2 += sh[8 + w]; }
    const float inv  = 1.0f / (float)(SLEN * HDIM);
    const float mean = S1 * inv;
    const float var  = S2 * inv - mean * mean;
    stats[bh * 2]     = mean;
    stats[bh * 2 + 1] = rsqrtf(var + 1e-5f);
  }
}

// ---------------------------------------------------------------------------
// normed = ((ao-mean)*rstd*gw + gb) * (1-0.8), relayout (B,H,S,D)->(B,S,H*D)
// ---------------------------------------------------------------------------
__global__ void norm_kernel(const float* __restrict__ AO, const float* __restrict__ stats,
                            const float* __restrict__ gw, const float* __restrict__ gb,
                            float* __restrict__ out) {
  const size_t total = (size_t)BATCH * NHEAD * SLEN * HDIM;
  for (size_t idx = (size_t)blockIdx.x * blockDim.x + threadIdx.x; idx < total;
       idx += (size_t)gridDim.x * blockDim.x) {
    const int d = (int)(idx & 63);
    const int s = (int)((idx >> 6) & (SLEN - 1));
    const int h = (int)((idx >> 17) & 15);
    const int b = (int)(idx >> 21);
    const int bh = b * NHEAD + h;
    const float mean = stats[bh * 2];
    const float rstd = stats[bh * 2 + 1];
    const float v = (AO[idx] - mean) * rstd * gw[h * HDIM + d] + gb[h * HDIM + d];
    out[((size_t)(b * SLEN + s) << 10) + h * HDIM + d] = v * 0.2f;
  }
}

// ---------------------------------------------------------------------------
extern "C" void kernel_launch(void* const* d_in, const int* in_sizes, int n_in,
                              void* d_out, int out_size, void* d_ws, size_t ws_size,
                              hipStream_t stream) {
  (void)in_sizes; (void)n_in; (void)out_size; (void)ws_size;
  const float* X    = (const float*)d_in[0];
  const float* mask = (const float*)d_in[1];
  const float* Wq   = (const float*)d_in[2];
  const float* Wk   = (const float*)d_in[3];
  const float* Wv   = (const float*)d_in[4];
  const float* Wo   = (const float*)d_in[5];
  const float* lq1  = (const float*)d_in[6];
  const float* lq2  = (const float*)d_in[7];
  const float* lk1  = (const float*)d_in[8];
  const float* lk2  = (const float*)d_in[9];
  const float* gw   = (const float*)d_in[10];
  const float* gb   = (const float*)d_in[11];

  float* ws = (float*)d_ws;
  float* Q     = ws;               // 8,388,608 floats (B,H,S,128)
  float* Kbuf  = ws + 8388608;     // 8,388,608
  float* V     = ws + 16777216;    // 4,194,304 (B,H,S,64)
  float* AO    = ws + 20971520;    // 4,194,304 (B,H,S,64)
  float* stats = ws + 25165824;    // 64
  float* lam   = ws + 25165888;    // 1
  float* NORM  = ws;               // reuse Q region after attention

  lambda_kernel<<<1, 32, 0, stream>>>(lq1, lq2, lk1, lk2, lam);

  // Projections: M=4096, K=1024; grid = (M/16) * (N/512)
  gemm_wmma<<<1024, 256, 0, stream>>>(X, Wq, Q,    4096, 2048, 1024, 1);
  gemm_wmma<<<1024, 256, 0, stream>>>(X, Wk, Kbuf, 4096, 2048, 1024, 1);
  gemm_wmma<<<512,  256, 0, stream>>>(X, Wv, V,    4096, 1024, 1024, 2);

  // Attention: 32 (b,h) * 16 q-blocks, 8 waves (16-row q tiles) each
  attn_kernel<<<512, 256, 0, stream>>>(Q, Kbuf, V, mask, lam, AO);

  stats_kernel<<<32, 256, 0, stream>>>(AO, stats);
  norm_kernel<<<2048, 256, 0, stream>>>(AO, stats, gw, gb, NORM);

  // Output projection into d_out
  gemm_wmma<<<512, 256, 0, stream>>>(NORM, Wo, (float*)d_out, 4096, 1024, 1024, 0);
}